// BiLSTM_Parser_26362509263485
// MI455X (gfx1250) — compile-verified
//
#include <hip/hip_runtime.h>
#include <hip/hip_bf16.h>
#include <stddef.h>
#include <stdint.h>

// ---------------------------------------------------------------------------
// Model dims (from reference)
// ---------------------------------------------------------------------------
#define SEQ   512
#define WDIM  256
#define PDIM  64
#define IN0   (WDIM + PDIM)   // 320
#define Hh    256             // hidden per direction
#define G4    (4 * Hh)        // 1024 gates
#define LSTMH 512             // 2 * Hh
#define MLPD  512

typedef float v2f __attribute__((ext_vector_type(2)));
typedef float v8f __attribute__((ext_vector_type(8)));
typedef unsigned int u32x4 __attribute__((ext_vector_type(4)));
typedef int  i32x4 __attribute__((ext_vector_type(4)));
typedef int  i32x8 __attribute__((ext_vector_type(8)));

// Tensor Data Mover path — this toolchain exposes the 6-arg builtin:
//   (uint32x4 g0, int32x8 g1, int32x4 g2, int32x4 g3, int32x8 g4, i32 cpol)
#if defined(__AMDGCN__) && __has_builtin(__builtin_amdgcn_tensor_load_to_lds) && \
    __has_builtin(__builtin_amdgcn_s_wait_tensorcnt)
#define USE_TDM 1
#else
#define USE_TDM 0
#endif

__device__ __forceinline__ float sigmoidf_(float x) {
    return 1.0f / (1.0f + __expf(-x));
}

// ---------------------------------------------------------------------------
// 1) Embedding gather
// ---------------------------------------------------------------------------
__global__ __launch_bounds__(256)
void k_embed(const int* __restrict__ wt, const int* __restrict__ pt,
             const float* __restrict__ we, const float* __restrict__ pe,
             float* __restrict__ x0)
{
    int idx = blockIdx.x * blockDim.x + threadIdx.x;
    if (idx >= SEQ * IN0) return;
    int t = idx / IN0;
    int c = idx - t * IN0;
    float v;
    if (c < WDIM) v = we[(size_t)wt[t] * WDIM + c];
    else          v = pe[(size_t)pt[t] * PDIM + (c - WDIM)];
    x0[idx] = v;
}

// ---------------------------------------------------------------------------
// 2) Transpose Whh (1024 x 256) -> WhhT (256 x 1024) for coalesced recurrence
// ---------------------------------------------------------------------------
__global__ __launch_bounds__(256)
void k_transpose_whh(const float* __restrict__ W, float* __restrict__ WT)
{
    int idx = blockIdx.x * blockDim.x + threadIdx.x;   // over 256*1024
    if (idx >= Hh * G4) return;
    int n = idx & (G4 - 1);
    int k = idx >> 10;
    WT[idx] = W[(size_t)n * Hh + k];
}

// ---------------------------------------------------------------------------
// 3) WMMA fp32 GEMM:  C[M][N] = A[M][K] @ W[N][K]^T (+ bias0[n] + bias1[n])
//    One wave computes TWO adjacent 16x16 C tiles (shared A fragment) with
//    V_WMMA_F32_16X16X4_F32. K and the W row stride are compile-time so the
//    inner loop is branch-free; fragments for k+4 are preloaded before the
//    k-step WMMAs issue (software pipeline -> loads overlap matrix math).
//
//    Fragment layouts per ISA 7.12.2 (32-bit A 16x4 / B 4x16 / C 16x16):
//      A: lanes 0-15 {A[m][k],A[m][k+1]}, lanes 16-31 {A[m][k+2],A[m][k+3]}
//      B: lanes 0-15 {W[n][k],W[n][k+1]}, lanes 16-31 {W[n][k+2],W[n][k+3]}
//      C: VGPR r -> row tm+r (lanes 0-15) / tm+r+8 (lanes 16-31), col tn+(lane&15)
// ---------------------------------------------------------------------------
template<int K, int LDW>
__global__ __launch_bounds__(32)
void k_gemm_xwT_wmma(const float* __restrict__ A, const float* __restrict__ W,
                     const float* __restrict__ bias0, const float* __restrict__ bias1,
                     float* __restrict__ C, int N)
{
    const int lane = threadIdx.x;       // full wave, EXEC all ones
    const int half = lane >> 4;         // 0 -> K pair {0,1}, 1 -> {2,3}
    const int l16  = lane & 15;
    const int tm   = blockIdx.y * 16;
    const int tn   = blockIdx.x * 32;   // two 16-wide N tiles per wave

    const float* arow = A + (size_t)(tm + l16)      * K   + half * 2;
    const float* wr0  = W + (size_t)(tn + l16)      * LDW + half * 2;
    const float* wr1  = W + (size_t)(tn + 16 + l16) * LDW + half * 2;

    v8f acc0 = {0.f, 0.f, 0.f, 0.f, 0.f, 0.f, 0.f, 0.f};
    v8f acc1 = {0.f, 0.f, 0.f, 0.f, 0.f, 0.f, 0.f, 0.f};

    v2f a, b0, b1;
    a.x  = arow[0]; a.y  = arow[1];
    b0.x = wr0[0];  b0.y = wr0[1];
    b1.x = wr1[0];  b1.y = wr1[1];

    #pragma unroll 8
    for (int k = 0; k < K - 4; k += 4) {
        v2f an, bn0, bn1;                      // preload next fragments
        an.x  = arow[k + 4]; an.y  = arow[k + 5];
        bn0.x = wr0[k + 4];  bn0.y = wr0[k + 5];
        bn1.x = wr1[k + 4];  bn1.y = wr1[k + 5];
        acc0 = __builtin_amdgcn_wmma_f32_16x16x4_f32(false, a, false, b0,
                                                     (short)0, acc0, false, false);
        acc1 = __builtin_amdgcn_wmma_f32_16x16x4_f32(false, a, false, b1,
                                                     (short)0, acc1, false, false);
        a = an; b0 = bn0; b1 = bn1;
    }
    acc0 = __builtin_amdgcn_wmma_f32_16x16x4_f32(false, a, false, b0,
                                                 (short)0, acc0, false, false);
    acc1 = __builtin_amdgcn_wmma_f32_16x16x4_f32(false, a, false, b1,
                                                 (short)0, acc1, false, false);

    const int cn0 = tn + l16;
    const int cn1 = tn + 16 + l16;
    float bs0 = 0.f, bs1 = 0.f;
    if (bias0) { bs0 += bias0[cn0]; bs1 += bias0[cn1]; }
    if (bias1) { bs0 += bias1[cn0]; bs1 += bias1[cn1]; }

    #pragma unroll
    for (int r = 0; r < 8; ++r) {
        int cm = tm + r + half * 8;
        C[(size_t)cm * N + cn0] = acc0[r] + bs0;
        C[(size_t)cm * N + cn1] = acc1[r] + bs1;
    }
}

// ---------------------------------------------------------------------------
// 4) LSTM scan (sequential over time). One block per direction (grid.x = 2).
// ---------------------------------------------------------------------------
__global__ __launch_bounds__(1024)
void k_lstm_scan2(const float* __restrict__ Gf, const float* __restrict__ WTf,
                  const float* __restrict__ Gb, const float* __restrict__ WTb,
                  float* __restrict__ out /* SEQ x 512 */)
{
    __shared__ float h_buf[2][Hh];
    __shared__ float gate[G4];

    const int  n   = threadIdx.x;
    const bool bwd = (blockIdx.x == 1);
    const float* G  = bwd ? Gb  : Gf;
    const float* WT = bwd ? WTb : WTf;
    const int  col  = bwd ? Hh : 0;

    if (n < Hh) h_buf[0][n] = 0.f;
    float c = 0.f;
    __syncthreads();

    int cur = 0;
    for (int s = 0; s < SEQ; ++s) {
        const int t = bwd ? (SEQ - 1 - s) : s;

        float acc = G[(size_t)t * G4 + n];
        const float* w = WT + n;
        #pragma unroll 8
        for (int k = 0; k < Hh; ++k)
            acc = fmaf(h_buf[cur][k], w[(size_t)k * G4], acc);
        gate[n] = acc;
        __syncthreads();

        if (n < Hh) {
            float ig = sigmoidf_(gate[n]);
            float fg = sigmoidf_(gate[Hh + n]);
            float gg = tanhf(gate[2 * Hh + n]);
            float og = sigmoidf_(gate[3 * Hh + n]);
            c = fg * c + ig * gg;
            float h = og * tanhf(c);
            h_buf[cur ^ 1][n] = h;
            out[(size_t)t * LSTMH + col + n] = h;
        }
        __syncthreads();
        cur ^= 1;
    }
}

// ---------------------------------------------------------------------------
// TDM helper: issue a TENSOR_LOAD_TO_LDS for a rows x cols f32 tile whose
// global row stride is stride_elems, destination LDS rows padded to cols+1
// floats via the D# pad feature (pad_interval code 6 = every 128 DWORDs,
// pad_amount 0 = 1 DWORD).  D# bitfields per CDNA5 ISA.
// ---------------------------------------------------------------------------
__device__ __forceinline__ void tdm_load_rows_f32(const float* gsrc, unsigned lds_byte,
                                                  int rows, int cols, int stride_elems)
{
#if USE_TDM
    unsigned long long ga = (unsigned long long)(uintptr_t)gsrc;
    u32x4 g0;
    g0[0] = 1u;                                     // count=1, user descriptor
    g0[1] = lds_byte;                               // lds_addr
    g0[2] = (unsigned)(ga & 0xffffffffu);           // global_addr[31:0]
    g0[3] = (unsigned)((ga >> 32) & 0x01ffffffu)    // global_addr[56:32]
            | (2u << 30);                           // type = 2 ("image")
    i32x8 g1;
    g1[0] = (int)((2u << 16)                        // data_size = 4 bytes
                | (1u << 20)                        // pad_enable
                | (6u << 22)                        // pad_interval: 128 DWORDs
                | (0u << 25));                      // pad_amount: 1 DWORD
    unsigned td0 = (unsigned)stride_elems;          // tensor_dim0
    unsigned td1 = (unsigned)rows;                  // tensor_dim1
    g1[1] = (int)((td0 & 0xffffu) << 16);                               // dim0[15:0]
    g1[2] = (int)(((td0 >> 16) & 0xffffu) | ((td1 & 0xffffu) << 16));   // dim0[31:16] | dim1[15:0]
    g1[3] = (int)(((td1 >> 16) & 0xffffu) | (((unsigned)cols) << 16));  // dim1[31:16] | tile_dim0
    g1[4] = (int)((unsigned)rows & 0xffffu);                            // tile_dim1 (tile_dim2=0)
    g1[5] = (int)(unsigned)stride_elems;                                // tensor_dim0_stride[31:0]
    g1[6] = 0;                                                          // stride hi / dim1_stride lo
    g1[7] = 0;
    i32x4 z4 = {0, 0, 0, 0};
    i32x8 z8 = {0, 0, 0, 0, 0, 0, 0, 0};
    // 6-arg form on this toolchain: (g0, g1, g2, g3, g4, cpol)
    __builtin_amdgcn_tensor_load_to_lds(g0, g1, z4, z4, z8, 0);
#else
    (void)gsrc; (void)lds_byte; (void)rows; (void)cols; (void)stride_elems;
#endif
}

// ---------------------------------------------------------------------------
// 5) Fused biaffine score:
//    S[i][j] = sum_k tanh(U[i][k] + V[j][k] + b_lin[k]) * w_out[k] + b_out
//    16x16 (i,j) tile per 256-thread block; k chunked through LDS.
//    U/V tile staging goes through the Tensor Data Mover: each of the 8 waves
//    issues one 2-row descriptor per matrix (uniform per-wave, no divergent
//    branch around the TDM op since TDM ignores EXEC), then waits on
//    TENSORcnt before the block barrier.
// ---------------------------------------------------------------------------
#define KCH 128
__global__ __launch_bounds__(256)
void k_biaffine(const float* __restrict__ U, const float* __restrict__ V,
                const float* __restrict__ b_lin, const float* __restrict__ w_out,
                const float* __restrict__ b_out, float* __restrict__ S)
{
    __shared__ float us[16][KCH + 1];   // +1 pad kills stride-128 bank conflicts
    __shared__ float vs[16][KCH + 1];
    __shared__ float bs[KCH];
    __shared__ float ws[KCH];

    const int tx = threadIdx.x, ty = threadIdx.y;
    const int tid = ty * 16 + tx;
    const int wv  = tid >> 5;                   // wave id 0..7
    const int i = blockIdx.y * 16 + ty;
    const int j = blockIdx.x * 16 + tx;

#if USE_TDM
    // Generic->LDS: flat shared addresses carry the LDS byte offset in the
    // low 32 bits (aperture base occupies addr[63:32]).
    const unsigned us_base = (unsigned)(uintptr_t)&us[0][0];
    const unsigned vs_base = (unsigned)(uintptr_t)&vs[0][0];
#endif

    float acc = 0.f;
    for (int kb = 0; kb < MLPD; kb += KCH) {
#if USE_TDM
        // Each wave DMAs rows [wv*2, wv*2+2) of both tiles into padded LDS.
        tdm_load_rows_f32(U + (size_t)(blockIdx.y * 16 + wv * 2) * MLPD + kb,
                          us_base + (unsigned)(wv * 2 * (KCH + 1) * 4),
                          2, KCH, MLPD);
        tdm_load_rows_f32(V + (size_t)(blockIdx.x * 16 + wv * 2) * MLPD + kb,
                          vs_base + (unsigned)(wv * 2 * (KCH + 1) * 4),
                          2, KCH, MLPD);
        if (tid < KCH) { bs[tid] = b_lin[kb + tid]; ws[tid] = w_out[kb + tid]; }
        __builtin_amdgcn_s_wait_tensorcnt(0);   // per-wave TDM completion
        __syncthreads();
#else
        #pragma unroll
        for (int e = 0; e < (16 * KCH) / 256; ++e) {
            int idx = tid + e * 256;
            int row = idx >> 7;
            int kk  = idx & (KCH - 1);
            us[row][kk] = U[(size_t)(blockIdx.y * 16 + row) * MLPD + kb + kk];
            vs[row][kk] = V[(size_t)(blockIdx.x * 16 + row) * MLPD + kb + kk];
        }
        if (tid < KCH) { bs[tid] = b_lin[kb + tid]; ws[tid] = w_out[kb + tid]; }
        __syncthreads();
#endif

        #pragma unroll 4
        for (int k = 0; k < KCH; ++k) {
            float tval = tanhf(us[ty][k] + vs[tx][k] + bs[k]);
            acc = fmaf(tval, ws[k], acc);
        }
        __syncthreads();
    }

    float r = acc + b_out[0];
    if (i == j) r = 0.f;
    S[(size_t)i * SEQ + j] = r;
}

// ---------------------------------------------------------------------------
// Launch
// ---------------------------------------------------------------------------
extern "C" void kernel_launch(void* const* d_in, const int* in_sizes, int n_in,
                              void* d_out, int out_size, void* d_ws, size_t ws_size,
                              hipStream_t stream)
{
    const int*   word_t  = (const int*)  d_in[0];
    const int*   pos_t   = (const int*)  d_in[1];
    const float* word_e  = (const float*)d_in[2];
    const float* pos_e   = (const float*)d_in[3];
    const float* wih0    = (const float*)d_in[4];   // (2,1024,320)
    const float* whh0    = (const float*)d_in[5];   // (2,1024,256)
    const float* bih0    = (const float*)d_in[6];   // (2,1024)
    const float* bhh0    = (const float*)d_in[7];
    const float* wih1    = (const float*)d_in[8];   // (2,1024,512)
    const float* whh1    = (const float*)d_in[9];
    const float* bih1    = (const float*)d_in[10];
    const float* bhh1    = (const float*)d_in[11];
    const float* W_lin   = (const float*)d_in[12];  // (512,1024)
    const float* b_lin   = (const float*)d_in[13];  // (512)
    const float* w_out   = (const float*)d_in[14];  // (1,512)
    const float* b_out   = (const float*)d_in[15];  // (1)
    float* S = (float*)d_out;                       // (512,512)

    float* ws = (float*)d_ws;
    size_t off = 0;
    float* x0   = ws + off; off += (size_t)SEQ * IN0;
    float* G0f  = ws + off; off += (size_t)SEQ * G4;
    float* G0b  = ws + off; off += (size_t)SEQ * G4;
    float* x1   = ws + off; off += (size_t)SEQ * LSTMH;
    float* G1f  = ws + off; off += (size_t)SEQ * G4;
    float* G1b  = ws + off; off += (size_t)SEQ * G4;
    float* x2   = ws + off; off += (size_t)SEQ * LSTMH;
    float* Umat = ws + off; off += (size_t)SEQ * MLPD;
    float* Vmat = ws + off; off += (size_t)SEQ * MLPD;
    float* WT0f = ws + off; off += (size_t)Hh * G4;
    float* WT0b = ws + off; off += (size_t)Hh * G4;
    float* WT1f = ws + off; off += (size_t)Hh * G4;
    float* WT1b = ws + off; off += (size_t)Hh * G4;

    // 1) embedding gather
    {
        int n = SEQ * IN0;
        k_embed<<<(n + 255) / 256, 256, 0, stream>>>(word_t, pos_t, word_e, pos_e, x0);
    }

    // 2) transpose recurrent weights
    {
        int n = Hh * G4;
        dim3 g((n + 255) / 256);
        k_transpose_whh<<<g, 256, 0, stream>>>(whh0,                 WT0f);
        k_transpose_whh<<<g, 256, 0, stream>>>(whh0 + (size_t)G4*Hh, WT0b);
        k_transpose_whh<<<g, 256, 0, stream>>>(whh1,                 WT1f);
        k_transpose_whh<<<g, 256, 0, stream>>>(whh1 + (size_t)G4*Hh, WT1b);
    }

    // 3) layer-0 gate pre-projections (K=320)
    {
        dim3 grid(G4 / 32, SEQ / 16);
        k_gemm_xwT_wmma<IN0, IN0><<<grid, 32, 0, stream>>>(x0, wih0,                  bih0,      bhh0,      G0f, G4);
        k_gemm_xwT_wmma<IN0, IN0><<<grid, 32, 0, stream>>>(x0, wih0 + (size_t)G4*IN0, bih0 + G4, bhh0 + G4, G0b, G4);
    }

    // 4) layer-0 recurrence (fwd + bwd concurrently)
    k_lstm_scan2<<<2, 1024, 0, stream>>>(G0f, WT0f, G0b, WT0b, x1);

    // 5) layer-1 gate pre-projections (K=512)
    {
        dim3 grid(G4 / 32, SEQ / 16);
        k_gemm_xwT_wmma<LSTMH, LSTMH><<<grid, 32, 0, stream>>>(x1, wih1,                    bih1,      bhh1,      G1f, G4);
        k_gemm_xwT_wmma<LSTMH, LSTMH><<<grid, 32, 0, stream>>>(x1, wih1 + (size_t)G4*LSTMH, bih1 + G4, bhh1 + G4, G1b, G4);
    }

    // 6) layer-1 recurrence
    k_lstm_scan2<<<2, 1024, 0, stream>>>(G1f, WT1f, G1b, WT1b, x2);

    // 7) biaffine projections: U = x2 @ W1.T, V = x2 @ W2.T (W row stride 1024)
    {
        dim3 grid(MLPD / 32, SEQ / 16);
        k_gemm_xwT_wmma<LSTMH, 2 * LSTMH><<<grid, 32, 0, stream>>>(x2, W_lin,         nullptr, nullptr, Umat, MLPD);
        k_gemm_xwT_wmma<LSTMH, 2 * LSTMH><<<grid, 32, 0, stream>>>(x2, W_lin + LSTMH, nullptr, nullptr, Vmat, MLPD);
    }

    // 8) fused tanh-biaffine scores with zeroed diagonal
    {
        dim3 grid(SEQ / 16, SEQ / 16);
        dim3 block(16, 16);
        k_biaffine<<<grid, block, 0, stream>>>(Umat, Vmat, b_lin, w_out, b_out, S);
    }
}